// PolyConv_25426206392749
// MI455X (gfx1250) — compile-verified
//
#include <hip/hip_runtime.h>

#define NN 100000
#define NE 1600000
#define IF 32
#define OF 64

typedef __attribute__((ext_vector_type(2))) float v2f;
typedef __attribute__((ext_vector_type(8))) float v8f;

// ---------------- elementwise helpers ----------------

__global__ void zero_kernel(float* __restrict__ p, int n) {
    int i = blockIdx.x * blockDim.x + threadIdx.x;
    if (i < n) p[i] = 0.0f;
}

// One thread per edge: sign mask + three degree scatter-adds.
__global__ void degree_kernel(const int* __restrict__ src, const int* __restrict__ dst,
                              const int* __restrict__ labels,
                              float* __restrict__ deg_o, float* __restrict__ deg_p,
                              float* __restrict__ deg_n, float* __restrict__ posmask,
                              int nEdges) {
    int e = blockIdx.x * blockDim.x + threadIdx.x;
    if (e >= nEdges) return;
    int s = src[e], d = dst[e];
    float pos = (labels[s] == labels[d]) ? 1.0f : 0.0f;
    posmask[e] = pos;
    atomicAdd(&deg_o[d], 1.0f);
    if (pos != 0.0f) atomicAdd(&deg_p[d], 1.0f);
    else             atomicAdd(&deg_n[d], 1.0f);
}

// In-place deg -> clip(deg,1)^-0.5 over the 3*NN degree block.
__global__ void dinv_kernel(float* __restrict__ d, int n) {
    int i = blockIdx.x * blockDim.x + threadIdx.x;
    if (i >= n) return;
    float x = d[i];
    x = x < 1.0f ? 1.0f : x;
    d[i] = rsqrtf(x);
}

// h[node*128 + colOff + j] = alpha * f[node*32 + j]
__global__ void seth_kernel(float* __restrict__ h, const float* __restrict__ f,
                            float alpha, int colOff, int n) {
    int i = blockIdx.x * blockDim.x + threadIdx.x;
    if (i >= n) return;
    int node = i >> 5, j = i & 31;
    h[node * 128 + colOff + j] = alpha * f[i];
}

// One wave32 per edge; lane = feature. src/dst/mask are wave-uniform ->
// gather f[src*32+lane] and atomic-add agg[dst*32+lane] are 128B coalesced.
__global__ void edge_scatter(const int* __restrict__ src, const int* __restrict__ dst,
                             const float* __restrict__ f, const float* __restrict__ dinv,
                             const float* __restrict__ posmask, int maskMode,
                             float* __restrict__ agg, int nEdges) {
    int t = blockIdx.x * blockDim.x + threadIdx.x;
    int e = t >> 5;
    int j = t & 31;
    if (e >= nEdges) return;
    int s = src[e], d = dst[e];
    float w = dinv[s];
    if (maskMode == 1)      w *= posmask[e];
    else if (maskMode == 2) w *= (1.0f - posmask[e]);
    if (w != 0.0f) {
        atomicAdd(&agg[d * 32 + j], f[s * 32 + j] * w);
    }
}

// f -= agg * Dinv[node]; h[.., colOff..] += alpha * f_new; agg = 0 (ready for next hop)
__global__ void update_kernel(float* __restrict__ f, float* __restrict__ agg,
                              const float* __restrict__ dinv, float* __restrict__ h,
                              float alpha, int colOff, int n) {
    int i = blockIdx.x * blockDim.x + threadIdx.x;
    if (i >= n) return;
    int node = i >> 5, j = i & 31;
    float fn = f[i] - agg[i] * dinv[node];
    f[i] = fn;
    agg[i] = 0.0f;
    h[node * 128 + colOff + j] += alpha * fn;
}

// ---------------- fp32 WMMA GEMM: out[M,64] = act(A[M,K] @ W[K,64] + b) ----------------
// One wave per 16-row strip; 4 column tiles of 16 -> 4 v8f accumulators.
// A fragment (ISA 7.12.2, 32-bit A 16x4): lanes 0-15 hold K=k0,k0+1; lanes 16-31 K=k0+2,k0+3.
// B fragment mirrored: lane half selects which K pair it supplies for its column.
__global__ void gemm_wmma_f32(const float* __restrict__ A, const float* __restrict__ W,
                              const float* __restrict__ bias, float* __restrict__ out,
                              int M, int K, int relu) {
    const int lane = threadIdx.x & 31;
    const int wave = threadIdx.x >> 5;
    const int strip = blockIdx.x * (blockDim.x >> 5) + wave;
    if (strip * 16 >= M) return;               // wave-uniform exit: EXEC all-1s for WMMA
    const int m0   = strip * 16;
    const int half = lane >> 4;                // 0: lanes 0-15, 1: lanes 16-31
    const int l16  = lane & 15;

    v8f acc0 = {}, acc1 = {}, acc2 = {}, acc3 = {};
    const int m = m0 + l16;

    for (int k0 = 0; k0 < K; k0 += 4) {
        const int kb = k0 + half * 2;
        v2f a;
        a.x = A[m * K + kb];
        a.y = A[m * K + kb + 1];

        v2f b0, b1, b2, b3;
        b0.x = W[(kb    ) * OF + 0 * 16 + l16];
        b0.y = W[(kb + 1) * OF + 0 * 16 + l16];
        b1.x = W[(kb    ) * OF + 1 * 16 + l16];
        b1.y = W[(kb + 1) * OF + 1 * 16 + l16];
        b2.x = W[(kb    ) * OF + 2 * 16 + l16];
        b2.y = W[(kb + 1) * OF + 2 * 16 + l16];
        b3.x = W[(kb    ) * OF + 3 * 16 + l16];
        b3.y = W[(kb + 1) * OF + 3 * 16 + l16];

        acc0 = __builtin_amdgcn_wmma_f32_16x16x4_f32(false, a, false, b0, (short)0, acc0, false, false);
        acc1 = __builtin_amdgcn_wmma_f32_16x16x4_f32(false, a, false, b1, (short)0, acc1, false, false);
        acc2 = __builtin_amdgcn_wmma_f32_16x16x4_f32(false, a, false, b2, (short)0, acc2, false, false);
        acc3 = __builtin_amdgcn_wmma_f32_16x16x4_f32(false, a, false, b3, (short)0, acc3, false, false);
    }

    // C/D layout: VGPR r holds row m0 + r + half*8, col = ctile*16 + l16.
    v8f accs[4] = {acc0, acc1, acc2, acc3};
#pragma unroll
    for (int c = 0; c < 4; ++c) {
#pragma unroll
        for (int r = 0; r < 8; ++r) {
            int row = m0 + r + half * 8;
            int col = c * 16 + l16;
            float v = accs[c][r] + bias[col];
            if (relu) v = v > 0.0f ? v : 0.01f * v;
            out[row * OF + col] = v;
        }
    }
}

// ---------------- host ----------------

extern "C" void kernel_launch(void* const* d_in, const int* in_sizes, int n_in,
                              void* d_out, int out_size, void* d_ws, size_t ws_size,
                              hipStream_t stream) {
    (void)in_sizes; (void)n_in; (void)out_size; (void)ws_size;

    const float* feat     = (const float*)d_in[0];
    const int*   esrc     = (const int*)d_in[1];
    const int*   edst     = (const int*)d_in[2];
    const int*   labels   = (const int*)d_in[3];
    const float* W_lin    = (const float*)d_in[4];
    const float* b_lin    = (const float*)d_in[5];
    const float* W_lin1   = (const float*)d_in[6];
    const float* b_lin1   = (const float*)d_in[7];
    const float* W_transh = (const float*)d_in[8];
    const float* b_transh = (const float*)d_in[9];
    float* out = (float*)d_out;

    // Workspace layout (floats): Dinv[3N] | posmask[E] | f[N*32] | agg[N*32] | hs_o[N*128] | hs_pn[N*128]
    float* ws      = (float*)d_ws;
    float* Dinv    = ws;                       // [0..N) = full, [N..2N) = pos, [2N..3N) = neg
    float* posmask = Dinv + 3 * NN;
    float* f       = posmask + NE;
    float* agg     = f + NN * 32;
    float* hs_o    = agg + NN * 32;
    float* hs_pn   = hs_o + NN * 128;

    static const float THETAS[4][3] = {
        {0.50f, -0.40f, 0.10f},
        {0.25f,  0.50f, -0.25f},
        {0.10f,  0.30f, 0.60f},
        {0.05f, -0.20f, 0.80f}};

    const int NF       = NN * 32;
    const int nfBlocks = (NF + 255) / 256;
    const int eBlocksT = (NE + 255) / 256;                 // one thread / edge
    const int eBlocksW = (NE * 32 + 255) / 256;            // one wave / edge

    // degrees + agg start at zero
    zero_kernel<<<(3 * NN + 255) / 256, 256, 0, stream>>>(Dinv, 3 * NN);
    zero_kernel<<<nfBlocks, 256, 0, stream>>>(agg, NF);
    degree_kernel<<<eBlocksT, 256, 0, stream>>>(esrc, edst, labels,
                                                Dinv, Dinv + NN, Dinv + 2 * NN,
                                                posmask, NE);
    dinv_kernel<<<(3 * NN + 255) / 256, 256, 0, stream>>>(Dinv, 3 * NN);

    // one polynomial group: thetas [t0,t1) with carried-over f, writes hs cols colBase+...
    auto run_group = [&](const float* dinv, int maskMode, int t0, int t1,
                         float* hs, int colBase) {
        hipMemcpyAsync(f, feat, (size_t)NF * sizeof(float),
                       hipMemcpyDeviceToDevice, stream);
        for (int t = t0; t < t1; ++t) {
            int colOff = colBase + (t - t0) * 32;
            seth_kernel<<<nfBlocks, 256, 0, stream>>>(hs, f, THETAS[t][0], colOff, NF);
            for (int k = 1; k < 3; ++k) {
                edge_scatter<<<eBlocksW, 256, 0, stream>>>(esrc, edst, f, dinv,
                                                           posmask, maskMode, agg, NE);
                update_kernel<<<nfBlocks, 256, 0, stream>>>(f, agg, dinv, hs,
                                                            THETAS[t][k], colOff, NF);
            }
        }
    };

    run_group(Dinv,          0, 0, 4, hs_o,  0);   // full graph, all 4 thetas
    run_group(Dinv + NN,     1, 0, 2, hs_pn, 0);   // pos edges, thetas[:2]
    run_group(Dinv + 2 * NN, 2, 2, 4, hs_pn, 64);  // neg edges, thetas[2:]

    // dense layers on the fp32 WMMA path
    const int gBlocks = (NN / 16 + 7) / 8;         // 8 waves/block, 16 rows/wave; 100000 % 16 == 0
    gemm_wmma_f32<<<gBlocks, 256, 0, stream>>>(hs_o,  W_lin,    b_lin,    out,               NN, 128, 1);
    gemm_wmma_f32<<<gBlocks, 256, 0, stream>>>(hs_pn, W_lin1,   b_lin1,   out + NN * OF,     NN, 128, 1);
    gemm_wmma_f32<<<gBlocks, 256, 0, stream>>>(feat,  W_transh, b_transh, out + 2 * NN * OF, NN, 32,  0);
}